// AttentionOutRNNUnit_90537910599926
// MI455X (gfx1250) — compile-verified
//
#include <hip/hip_runtime.h>
#include <math.h>

typedef __attribute__((ext_vector_type(2))) float v2f;
typedef __attribute__((ext_vector_type(8))) float v8f;

#define B_ 64
#define L_ 4096
#define E_ 256
#define H_ 256

#define SPLIT 16
#define WAVES 8
#define TPB (WAVES * 32)                        // 256 threads = 8 wave32
#define ROWS_PER_BLOCK (L_ / SPLIT)             // 256
#define ROWS_PER_WAVE (ROWS_PER_BLOCK / WAVES)  // 32

// ---------------------------------------------------------------------------
// Kernel A: Wh[b,e] = sum_h W[e,h] * hidden[b,h]   (64x256x256 GEMM, fp32 WMMA)
// One wave per 16x16 output tile; V_WMMA_F32_16X16X4_F32, K-loop over H.
// A 16x4 layout: lanes 0-15 hold K={0,1}, lanes 16-31 hold K={2,3} (v2f).
// B 4x16 layout: row striped across lanes within a VGPR, same K split.
// D layout: VGPR j -> M=j (lanes 0-15) / M=8+j (lanes 16-31), N = lane&15.
// ---------------------------------------------------------------------------
__global__ __launch_bounds__(32)
void wh_wmma_kernel(const float* __restrict__ hidden,
                    const float* __restrict__ W,
                    float* __restrict__ Wh)
{
    const int tile = blockIdx.x;
    const int m0 = (tile & 3) * 16;      // over B (64/16 = 4 tiles)
    const int n0 = (tile >> 2) * 16;     // over E (256/16 = 16 tiles)
    const int lane = threadIdx.x;
    const int r  = lane & 15;
    const int kh = (lane >> 4) * 2;      // 0 or 2

    const float* ha = hidden + (m0 + r) * H_ + kh;  // A: hidden[b, h]
    const float* wb = W      + (n0 + r) * H_ + kh;  // B: W^T[h, e] = W[e, h]

    v8f c = {0.f, 0.f, 0.f, 0.f, 0.f, 0.f, 0.f, 0.f};
    for (int k0 = 0; k0 < H_; k0 += 4) {
        v2f a, b;
        a.x = ha[k0]; a.y = ha[k0 + 1];
        b.x = wb[k0]; b.y = wb[k0 + 1];
        c = __builtin_amdgcn_wmma_f32_16x16x4_f32(
                /*neg_a=*/false, a, /*neg_b=*/false, b,
                /*c_mod=*/(short)0, c, /*reuse_a=*/false, /*reuse_b=*/false);
    }

    const int mbase = m0 + (lane >> 4) * 8;
    #pragma unroll
    for (int j = 0; j < 8; ++j)
        Wh[(mbase + j) * E_ + n0 + r] = c[j];
}

// ---------------------------------------------------------------------------
// Kernel B: fused scores + online softmax + context accumulation.
// Grid (B, SPLIT); each wave streams 32 contiguous rows of enc[b] once,
// keeping running (max, sum, ctx[8] per lane). Waves combine in LDS;
// block writes one rescalable partial (m, s, ctx[256]) to workspace.
// ---------------------------------------------------------------------------
__global__ __launch_bounds__(TPB)
void attn_partial_kernel(const float* __restrict__ enc,
                         const float* __restrict__ Wh,
                         float* __restrict__ pm,
                         float* __restrict__ ps,
                         float* __restrict__ pctx)
{
    __shared__ float s_m[WAVES];
    __shared__ float s_s[WAVES];
    __shared__ float s_ctx[WAVES][E_];

    const int b    = blockIdx.x;
    const int spl  = blockIdx.y;
    const int tid  = threadIdx.x;
    const int wave = tid >> 5;
    const int lane = tid & 31;

    // Per-lane chunk of Wh[b,:]: 8 floats (32 lanes * 8 = 256 = E)
    const float* whp = Wh + b * E_ + lane * 8;
    const float4 w0 = *(const float4*)(whp);
    const float4 w1 = *(const float4*)(whp + 4);

    const int l0 = spl * ROWS_PER_BLOCK + wave * ROWS_PER_WAVE;
    const float* p = enc + ((size_t)b * L_ + (size_t)l0) * E_ + lane * 8;

    float m = -INFINITY;
    float s = 0.f;
    float acc[8] = {0.f, 0.f, 0.f, 0.f, 0.f, 0.f, 0.f, 0.f};

    for (int r = 0; r < ROWS_PER_WAVE; ++r) {
        const float* rp = p + (size_t)r * E_;
        if (r + 4 < ROWS_PER_WAVE)
            __builtin_prefetch(rp + 4 * E_, 0, 0);   // global_prefetch_b8

        const float4 x0 = *(const float4*)rp;
        const float4 x1 = *(const float4*)(rp + 4);

        float d = x0.x * w0.x + x0.y * w0.y + x0.z * w0.z + x0.w * w0.w
                + x1.x * w1.x + x1.y * w1.y + x1.z * w1.z + x1.w * w1.w;
        #pragma unroll
        for (int off = 16; off > 0; off >>= 1)
            d += __shfl_xor(d, off, 32);
        // d is now wave-uniform: score for row l0 + r (bias dropped:
        // a constant shift is softmax-invariant).

        if (d > m) {                       // wave-uniform branch, rare after warmup
            const float corr = __expf(m - d);  // exp(-inf)=0 on first hit
            s *= corr;
            #pragma unroll
            for (int j = 0; j < 8; ++j) acc[j] *= corr;
            m = d;
        }
        const float pr = __expf(d - m);
        s += pr;
        acc[0] += pr * x0.x; acc[1] += pr * x0.y;
        acc[2] += pr * x0.z; acc[3] += pr * x0.w;
        acc[4] += pr * x1.x; acc[5] += pr * x1.y;
        acc[6] += pr * x1.z; acc[7] += pr * x1.w;
    }

    #pragma unroll
    for (int j = 0; j < 8; ++j) s_ctx[wave][lane * 8 + j] = acc[j];
    if (lane == 0) { s_m[wave] = m; s_s[wave] = s; }
    __syncthreads();

    // Combine the 8 wave-partials; thread tid handles channel e = tid.
    float gm = s_m[0];
    #pragma unroll
    for (int w = 1; w < WAVES; ++w) gm = fmaxf(gm, s_m[w]);
    float gs = 0.f, gc = 0.f;
    #pragma unroll
    for (int w = 0; w < WAVES; ++w) {
        const float f = __expf(s_m[w] - gm);
        gs += s_s[w] * f;
        gc += s_ctx[w][tid] * f;
    }
    const int pidx = b * SPLIT + spl;
    pctx[(size_t)pidx * E_ + tid] = gc;
    if (tid == 0) { pm[pidx] = gm; ps[pidx] = gs; }
}

// ---------------------------------------------------------------------------
// Kernel C: combine SPLIT partials per batch -> context[b, e].
// ---------------------------------------------------------------------------
__global__ __launch_bounds__(E_)
void attn_final_kernel(const float* __restrict__ pm,
                       const float* __restrict__ ps,
                       const float* __restrict__ pctx,
                       float* __restrict__ out)
{
    const int b = blockIdx.x;
    const int e = threadIdx.x;

    float gm = -INFINITY;
    #pragma unroll
    for (int sp = 0; sp < SPLIT; ++sp)
        gm = fmaxf(gm, pm[b * SPLIT + sp]);

    float gs = 0.f, gc = 0.f;
    #pragma unroll
    for (int sp = 0; sp < SPLIT; ++sp) {
        const float f = __expf(pm[b * SPLIT + sp] - gm);
        gs += ps[b * SPLIT + sp] * f;
        gc += pctx[(size_t)(b * SPLIT + sp) * E_ + e] * f;
    }
    out[b * E_ + e] = gc / gs;
}

extern "C" void kernel_launch(void* const* d_in, const int* in_sizes, int n_in,
                              void* d_out, int out_size, void* d_ws, size_t ws_size,
                              hipStream_t stream) {
    (void)in_sizes; (void)n_in; (void)out_size; (void)ws_size;

    const float* hidden = (const float*)d_in[0];   // [B, H]
    const float* enc    = (const float*)d_in[1];   // [B, L, E]
    const float* W      = (const float*)d_in[2];   // [E, H]
    // d_in[3] = bias[1]: softmax-invariant, intentionally unused.
    float* out = (float*)d_out;                    // [B, E]

    // Workspace layout (floats): Wh[B*E] | pm[B*SPLIT] | ps[B*SPLIT] | pctx[B*SPLIT*E]
    float* Wh   = (float*)d_ws;
    float* pm   = Wh + B_ * E_;
    float* ps   = pm + B_ * SPLIT;
    float* pctx = ps + B_ * SPLIT;

    wh_wmma_kernel<<<(B_ / 16) * (E_ / 16), 32, 0, stream>>>(hidden, W, Wh);
    attn_partial_kernel<<<dim3(B_, SPLIT), TPB, 0, stream>>>(enc, Wh, pm, ps, pctx);
    attn_final_kernel<<<B_, E_, 0, stream>>>(pm, ps, pctx, out);
}